// KVQuantizerDequantizer_76407468195935
// MI455X (gfx1250) — compile-verified
//
#include <hip/hip_runtime.h>
#include <stdint.h>

// Groupwise 4-bit asymmetric RTN quantize + dequantize, groupsize = 128 along
// the contiguous last dim. One wave32 per group: lane l owns elements
// [4l, 4l+4) (float4), wave-level shfl_xor min/max reduction, then affine
// quant/dequant. Data path: async global->LDS b128 double buffering
// (ASYNCcnt) + nontemporal b128 stores. Memory-bound: 512MB @ 23.3TB/s.

typedef float v4f __attribute__((ext_vector_type(4)));

#define QGROUP 128
#define MAXQ 15.0f
#define QEPS 1e-8f
#define CH 4                         // groups staged per tile per wave
#define WAVES_PER_BLOCK 8
#define BLOCK_THREADS (WAVES_PER_BLOCK * 32)
#define TILE_FLOATS (CH * QGROUP)    // 512 floats = 2KB per buffer

__global__ __launch_bounds__(BLOCK_THREADS)
void kvq_rtn_dq_kernel(const float* __restrict__ x,
                       float* __restrict__ out,
                       int ngroups)
{
    // Per-wave private double buffer: 8 waves * 2 * 2KB = 32KB LDS / block.
    __shared__ __align__(16) float smem[WAVES_PER_BLOCK][2][TILE_FLOATS];

    const int lane = threadIdx.x & 31;
    const int wib  = threadIdx.x >> 5;                       // wave in block
    const int wid  = blockIdx.x * WAVES_PER_BLOCK + wib;     // global wave id
    const int nw   = gridDim.x * WAVES_PER_BLOCK;            // total waves
    const int ntiles = (ngroups + CH - 1) / CH;

    // Issue CH async b128 loads (one group each: 32 lanes x 16B = 512B).
    // Group index clamped in-range so we always issue exactly CH ops
    // (keeps the s_wait_asynccnt immediates constant); garbage-staged
    // groups are never processed.
    auto issue_tile = [&](int t, float* dst) {
#pragma unroll
        for (int i = 0; i < CH; ++i) {
            int g  = t * CH + i;
            int gc = (g < ngroups) ? g : (ngroups - 1);
            uint64_t gaddr = (uint64_t)(uintptr_t)(x + (size_t)gc * QGROUP + lane * 4);
            uint32_t ldsa  = (uint32_t)(uintptr_t)(dst + i * QGROUP + lane * 4);
            asm volatile("global_load_async_to_lds_b128 %0, %1, off"
                         :: "v"(ldsa), "v"(gaddr)
                         : "memory");
        }
    };

    int parity = 0;
    if (wid < ntiles) issue_tile(wid, &smem[wib][0][0]);

    for (int t = wid; t < ntiles; t += nw) {
        int tn = t + nw;
        if (tn < ntiles) {
            // Prefetch next tile into the other buffer, then wait for the
            // CURRENT tile: CH new ops outstanding, so asynccnt <= CH means
            // the previous CH (current tile) completed (in-order completion).
            issue_tile(tn, &smem[wib][parity ^ 1][0]);
            asm volatile("s_wait_asynccnt 0x4" ::: "memory");
        } else {
            asm volatile("s_wait_asynccnt 0x0" ::: "memory");
        }

        const float* cur = &smem[wib][parity][0];
#pragma unroll
        for (int i = 0; i < CH; ++i) {
            int g = t * CH + i;
            if (g < ngroups) {
                // ds_load_b128: lane l holds elements [4l, 4l+4) of the group
                v4f v = *(const v4f*)(cur + i * QGROUP + lane * 4);

                float mn = fminf(fminf(v.x, v.y), fminf(v.z, v.w));
                float mx = fmaxf(fmaxf(v.x, v.y), fmaxf(v.z, v.w));
                // wave32 butterfly reduction (5 steps)
#pragma unroll
                for (int off = 16; off > 0; off >>= 1) {
                    mn = fminf(mn, __shfl_xor(mn, off, 32));
                    mx = fmaxf(mx, __shfl_xor(mx, off, 32));
                }

                float scale = fmaxf((mx - mn) / MAXQ, QEPS);
                float zp    = rintf(-mn / scale);   // offset (round half-even)

                v4f r;
                r.x = (fminf(fmaxf(rintf(v.x / scale) + zp, 0.0f), MAXQ) - zp) * scale;
                r.y = (fminf(fmaxf(rintf(v.y / scale) + zp, 0.0f), MAXQ) - zp) * scale;
                r.z = (fminf(fmaxf(rintf(v.z / scale) + zp, 0.0f), MAXQ) - zp) * scale;
                r.w = (fminf(fmaxf(rintf(v.w / scale) + zp, 0.0f), MAXQ) - zp) * scale;

                // Write-once output: NT store, don't thrash L2.
                __builtin_nontemporal_store(r, (v4f*)(out + (size_t)g * QGROUP + lane * 4));
            }
        }
        parity ^= 1;
    }
}

extern "C" void kernel_launch(void* const* d_in, const int* in_sizes, int n_in,
                              void* d_out, int out_size, void* d_ws, size_t ws_size,
                              hipStream_t stream) {
    (void)n_in; (void)out_size; (void)d_ws; (void)ws_size;
    const float* x = (const float*)d_in[0];
    float* out = (float*)d_out;

    int n = in_sizes[0];
    int ngroups = n / QGROUP;                       // 524288 for the ref shape
    int ntiles  = (ngroups + CH - 1) / CH;          // 131072

    int blocks = (ntiles + WAVES_PER_BLOCK - 1) / WAVES_PER_BLOCK;
    if (blocks > 4096) blocks = 4096;               // ~4 tiles per wave
    if (blocks < 1) blocks = 1;

    kvq_rtn_dq_kernel<<<blocks, BLOCK_THREADS, 0, stream>>>(x, out, ngroups);
}